// SpatialAffineTransformer_54065048322998
// MI455X (gfx1250) — compile-verified
//
#include <hip/hip_runtime.h>
#include <hip/hip_bf16.h>

typedef __attribute__((ext_vector_type(2))) float v2f;
typedef __attribute__((ext_vector_type(8))) float v8f;

#define BATCH 16
#define HI 384
#define WI 384
#define CH 32
#define HW (HI * WI)                 // 147456
#define SLICES 64
#define HW_PER_SLICE (HW / SLICES)   // 2304
#define OUT_H 384
#define OUT_W 384
#define NPIX (OUT_H * OUT_W)

// ---------------------------------------------------------------------------
// Kernel 1: per-(batch, slice) partial channel sums.
// grid = (SLICES, BATCH), block = 256.  Each wave reads 512B contiguous per
// iteration (8 lanes x float4 = one 128B pixel, 4 pixels per wave).
// ---------------------------------------------------------------------------
__global__ void stn_pool_partial(const float* __restrict__ X,
                                 float* __restrict__ partials) {
    const int slice = blockIdx.x;
    const int b     = blockIdx.y;
    const int tid   = threadIdx.x;
    const int cq    = tid & 7;    // channel quad: channels cq*4 .. cq*4+3
    const int hg    = tid >> 3;   // 0..31 pixel offset within iteration group
    const float* img = X + (size_t)b * HW * CH;
    const int h0 = slice * HW_PER_SLICE;

    float4 acc = make_float4(0.f, 0.f, 0.f, 0.f);
#pragma unroll 4
    for (int i = 0; i < HW_PER_SLICE / 32; ++i) {
        const int h = h0 + i * 32 + hg;
        const float4 v =
            *reinterpret_cast<const float4*>(img + (size_t)h * CH + cq * 4);
        const int hp = h + 8 * 32;  // prefetch 8 iterations ahead
        if (hp < h0 + HW_PER_SLICE)
            __builtin_prefetch(img + (size_t)hp * CH + cq * 4, 0, 1);
        acc.x += v.x; acc.y += v.y; acc.z += v.z; acc.w += v.w;
    }

    __shared__ float4 red[256];
    red[tid] = acc;
    __syncthreads();
    if (tid < 8) {
        float4 s = make_float4(0.f, 0.f, 0.f, 0.f);
        for (int k = 0; k < 32; ++k) {
            const float4 v = red[k * 8 + tid];
            s.x += v.x; s.y += v.y; s.z += v.z; s.w += v.w;
        }
        float* dst = partials + ((size_t)(b * SLICES + slice)) * CH + tid * 4;
        dst[0] = s.x; dst[1] = s.y; dst[2] = s.z; dst[3] = s.w;
    }
}

// ---------------------------------------------------------------------------
// Kernel 2: finish mean, then theta = feat @ W_loc + b_loc via
// V_WMMA_F32_16X16X4_F32 (M=16 batches, N=16 padded from 6, K=32 as 8 chained
// K=4 steps).  One wave, EXEC all-ones through the WMMA chain.
//
// f32 A(16x4) lane layout (ISA 7.12.2): lane l holds M = l&15;
//   lanes 0-15: VGPR0=K0, VGPR1=K1; lanes 16-31: VGPR0=K2, VGPR1=K3.
// f32 C/D(16x16): VGPR j, lanes 0-15 -> (M=j, N=lane); lanes 16-31 -> (M=j+8).
// ---------------------------------------------------------------------------
__global__ void stn_theta_wmma(const float* __restrict__ partials,
                               const float* __restrict__ W_loc,
                               const float* __restrict__ b_loc,
                               float* __restrict__ theta) {
    __shared__ float feat[BATCH * CH];  // feat[b][c], 512 floats
    __shared__ float wpad[CH * 16];     // W_loc zero-padded to 32x16
    const int lane = threadIdx.x;       // 0..31

    // feat[b][c] = sum over slices / HW; iteration i handles batch b=i,
    // channel c=lane -> fully coalesced 128B reads of partials.
    for (int i = 0; i < BATCH; ++i) {
        float s = 0.f;
        for (int sl = 0; sl < SLICES; ++sl)
            s += partials[((size_t)(i * SLICES + sl)) * CH + lane];
        feat[i * CH + lane] = s * (1.0f / (float)HW);
    }
    // wpad[k][n] = W_loc[k][n] for n<6 else 0
    for (int i = 0; i < (CH * 16) / 32; ++i) {
        const int p = i * 32 + lane;
        const int k = p >> 4, n = p & 15;
        wpad[p] = (n < 6) ? W_loc[k * 6 + n] : 0.0f;
    }
    __syncthreads();

    const int half = lane >> 4;   // selects K pair within each K=4 step
    const int lrow = lane & 15;   // M index for A, N index for B/D
    v8f acc = {};
#pragma unroll
    for (int kk = 0; kk < 8; ++kk) {
        const int k0 = kk * 4 + half * 2;
        v2f a, bm;
        a.x  = feat[lrow * CH + k0];
        a.y  = feat[lrow * CH + k0 + 1];
        bm.x = wpad[k0 * 16 + lrow];
        bm.y = wpad[(k0 + 1) * 16 + lrow];
        acc = __builtin_amdgcn_wmma_f32_16x16x4_f32(
            /*neg_a=*/false, a, /*neg_b=*/false, bm,
            /*c_mod=*/(short)0, acc, /*reuse_a=*/false, /*reuse_b=*/false);
    }

    // D(m, n): m<8 -> acc[m] at lane n; m>=8 -> acc[m-8] at lane n+16.
    if (lrow < 6) {
        const float bias = b_loc[lrow];
#pragma unroll
        for (int j = 0; j < 8; ++j) {
            const int m = j + half * 8;
            theta[m * 6 + lrow] = acc[j] + bias;
        }
    }
}

// ---------------------------------------------------------------------------
// Kernel 3: affine grid + bilinear gather.  Each wave32 = 32 channels of one
// output pixel: all 4 taps and the store are single 128B coalesced requests.
// grid = (NPIX/8, BATCH), block = 256 (8 waves -> 8 pixels per block).
// ---------------------------------------------------------------------------
__global__ void stn_sampler(const float* __restrict__ X,
                            const float* __restrict__ theta,
                            float* __restrict__ out) {
    const int b = blockIdx.y;
    const int p = blockIdx.x * 8 + (threadIdx.x >> 5);
    const int c = threadIdx.x & 31;
    const int oy = p / OUT_W;
    const int ox = p - oy * OUT_W;

    const float gx = ox * (2.0f / (float)(OUT_W - 1)) - 1.0f;
    const float gy = oy * (2.0f / (float)(OUT_H - 1)) - 1.0f;

    const float* t = theta + b * 6;  // wave-uniform, L2-resident
    const float sx = t[0] * gx + t[1] * gy + t[2];
    const float sy = t[3] * gx + t[4] * gy + t[5];

    const float xp = (sx + 1.0f) * (0.5f * (float)(WI - 1));
    const float yp = (sy + 1.0f) * (0.5f * (float)(HI - 1));
    const float x0f = floorf(xp), y0f = floorf(yp);
    int x0 = (int)x0f, y0 = (int)y0f;
    const int x1 = min(max(x0 + 1, 0), WI - 1);
    const int y1 = min(max(y0 + 1, 0), HI - 1);
    x0 = min(max(x0, 0), WI - 1);
    y0 = min(max(y0, 0), HI - 1);
    const float wx1 = xp - x0f, wx0 = 1.0f - wx1;
    const float wy1 = yp - y0f, wy0 = 1.0f - wy1;

    const float* img = X + (size_t)b * HW * CH;
    const float v00 = img[((size_t)(y0 * WI + x0)) * CH + c];
    const float v01 = img[((size_t)(y0 * WI + x1)) * CH + c];
    const float v10 = img[((size_t)(y1 * WI + x0)) * CH + c];
    const float v11 = img[((size_t)(y1 * WI + x1)) * CH + c];

    out[((size_t)b * NPIX + p) * CH + c] =
        v00 * (wy0 * wx0) + v01 * (wy0 * wx1) +
        v10 * (wy1 * wx0) + v11 * (wy1 * wx1);
}

// ---------------------------------------------------------------------------
extern "C" void kernel_launch(void* const* d_in, const int* in_sizes, int n_in,
                              void* d_out, int out_size, void* d_ws,
                              size_t ws_size, hipStream_t stream) {
    (void)in_sizes; (void)n_in; (void)out_size; (void)ws_size;
    const float* X     = (const float*)d_in[0];
    const float* W_loc = (const float*)d_in[1];
    const float* b_loc = (const float*)d_in[2];
    float* out = (float*)d_out;

    // Workspace: [BATCH*SLICES*CH] partial sums (128 KB) + [96] theta floats.
    float* partials = (float*)d_ws;
    float* theta    = partials + BATCH * SLICES * CH;

    stn_pool_partial<<<dim3(SLICES, BATCH), 256, 0, stream>>>(X, partials);
    stn_theta_wmma<<<1, 32, 0, stream>>>(partials, W_loc, b_loc, theta);
    stn_sampler<<<dim3(NPIX / 8, BATCH), 256, 0, stream>>>(X, theta, out);
}